// TensorParallelMoeMLP_14233521619263
// MI455X (gfx1250) — compile-verified
//
#include <hip/hip_runtime.h>
#include <hip/hip_bf16.h>

// ---------------- problem constants ----------------
constexpr int NTOK  = 4096;          // B*T
constexpr int HD    = 2048;          // hidden
constexpr int ID    = 1408;          // intermediate
constexpr int NE    = 8;             // experts
constexpr int TOPK  = 2;
constexpr int ROWT  = 32;            // rows per WMMA tile (2 x 16)
constexpr int PAIR_CAP  = NTOK * TOPK + ROWT * NE;  // 8448 (32-aligned per-expert pad)
constexpr int MAX_TILES = PAIR_CAP / ROWT;          // 264

// ---------------- workspace layout (bytes) ----------------
constexpr size_t OFF_COUNT   = 0;                        // int[8]
constexpr size_t OFF_CURSOR  = 64;                       // int[8]
constexpr size_t OFF_PROBSUM = 128;                      // float[8]
constexpr size_t OFF_TILEEXP = 192;                      // int[MAX_TILES]
constexpr size_t OFF_TOPIDX  = 1280;                     // int[NTOK*2]
constexpr size_t OFF_TOPVAL  = 34048;                    // float[NTOK*2]
constexpr size_t OFF_PAIRTOK = 66816;                    // int[PAIR_CAP]
constexpr size_t OFF_PAIRW   = 100608;                   // float[PAIR_CAP]
constexpr size_t OFF_ZERO    = 134400;                   // float[HD] zero row
constexpr size_t OFF_H1      = 142592;                   // float[PAIR_CAP*ID] (~47.6 MB)
constexpr size_t OFF_HS      = OFF_H1 + (size_t)PAIR_CAP * ID * 4; // float[NTOK*ID] (~23.1 MB)

typedef __attribute__((ext_vector_type(2))) float v2f;
typedef __attribute__((ext_vector_type(8))) float v8f;

__device__ __forceinline__ v8f wmma4(v2f a, v2f b, v8f c) {
    // D = A(16x4) * B(4x16) + C, fp32 exact path
    return __builtin_amdgcn_wmma_f32_16x16x4_f32(
        /*neg_a=*/false, a, /*neg_b=*/false, b,
        /*c_mod=*/(short)0, c, /*reuse_a=*/false, /*reuse_b=*/false);
}

__device__ __forceinline__ float silu(float v) {
    // v * sigmoid(v) with HW approximate reciprocal (v_rcp_f32, ~1 ulp)
    return v * __builtin_amdgcn_rcpf(1.0f + __expf(-v));
}

// ---------------- init ----------------
__global__ void init_kernel(int* count, int* cursor, float* prob_sum, int* pair_tok,
                            float* zrow) {
    int i = blockIdx.x * 256 + threadIdx.x;
    if (i < NE) { count[i] = 0; cursor[i] = 0; prob_sum[i] = 0.f; }
    if (i < HD) zrow[i] = 0.f;
    if (i < PAIR_CAP) pair_tok[i] = -1;
}

// ---------------- router: wave per token ----------------
__global__ void router_kernel(const float* __restrict__ x, const float* __restrict__ gw,
                              int* count, float* prob_sum,
                              int* top_idx, float* top_val) {
    const int wave = threadIdx.x >> 5;
    const int lane = threadIdx.x & 31;
    const int t = blockIdx.x * 4 + wave;
    const float* xr = x + (size_t)t * HD;

    float acc[NE];
#pragma unroll
    for (int e = 0; e < NE; ++e) acc[e] = 0.f;

    for (int h = lane; h < HD; h += 32) {
        float xv = xr[h];
#pragma unroll
        for (int e = 0; e < NE; ++e)
            acc[e] = fmaf(xv, gw[e * HD + h], acc[e]);
    }
#pragma unroll
    for (int e = 0; e < NE; ++e)
        for (int off = 16; off >= 1; off >>= 1)
            acc[e] += __shfl_xor(acc[e], off, 32);

    float mx = acc[0];
#pragma unroll
    for (int e = 1; e < NE; ++e) mx = fmaxf(mx, acc[e]);
    float p[NE]; float s = 0.f;
#pragma unroll
    for (int e = 0; e < NE; ++e) { p[e] = __expf(acc[e] - mx); s += p[e]; }
    float inv = 1.0f / s;
#pragma unroll
    for (int e = 0; e < NE; ++e) p[e] *= inv;

    int i1 = 0; float v1 = p[0];
#pragma unroll
    for (int e = 1; e < NE; ++e) if (p[e] > v1) { v1 = p[e]; i1 = e; }
    int i2 = -1; float v2 = -1.f;
#pragma unroll
    for (int e = 0; e < NE; ++e) if (e != i1 && p[e] > v2) { v2 = p[e]; i2 = e; }

    if (lane == 0) {
        top_idx[t * 2 + 0] = i1;  top_idx[t * 2 + 1] = i2;
        top_val[t * 2 + 0] = v1;  top_val[t * 2 + 1] = v2;
        atomicAdd(&count[i1], 1); atomicAdd(&count[i2], 1);
#pragma unroll
        for (int e = 0; e < NE; ++e) atomicAdd(&prob_sum[e], p[e]);
    }
}

// ---------------- per-expert 32-aligned offsets + tile->expert map ----------------
__global__ void build_offsets(const int* count, int* cursor, int* tile_expert) {
    int off = 0;
    for (int e = 0; e < NE; ++e) {
        cursor[e] = off;
        int tiles = (count[e] + ROWT - 1) / ROWT;
        for (int t = 0; t < tiles; ++t) tile_expert[off / ROWT + t] = e;
        off += tiles * ROWT;
    }
    for (int t = off / ROWT; t < MAX_TILES; ++t) tile_expert[t] = -1;
}

__global__ void scatter_kernel(const int* __restrict__ top_idx, const float* __restrict__ top_val,
                               int* cursor, int* pair_tok, float* pair_w) {
    int t = blockIdx.x * 256 + threadIdx.x;
    if (t >= NTOK) return;
#pragma unroll
    for (int k = 0; k < TOPK; ++k) {
        int e = top_idx[t * 2 + k];
        int pos = atomicAdd(&cursor[e], 1);
        pair_tok[pos] = t;
        pair_w[pos] = top_val[t * 2 + k];
    }
}

__global__ void aux_kernel(const int* count, const float* prob_sum, float* out_aux) {
    float a = 0.f;
    for (int e = 0; e < NE; ++e) {
        float importance = prob_sum[e] / (float)NTOK;
        float load = (float)count[e] / (float)(NTOK * TOPK);
        a += importance * load;
    }
    out_aux[0] = a * (float)NE;
}

// ---------------- fc1: OUT[row, i] = SiLU( X[row,:] @ W[e][:, i] ) ----------------
// Each wave: 32 rows x 64 cols; B fragments reused by two row groups.
template <bool GATHER>
__global__ void fc1_kernel(const float* __restrict__ X, const float* __restrict__ W,
                           float* __restrict__ OUT,
                           const int* __restrict__ tile_expert,
                           const int* __restrict__ pair_tok,
                           const float* __restrict__ zrow) {
    __shared__ int s_tok[ROWT];
    const int tile = blockIdx.x;
    int e = 0;
    if constexpr (GATHER) {
        e = tile_expert[tile];
        if (e < 0) return;
    }
    if (threadIdx.x < ROWT) {
        int r = tile * ROWT + threadIdx.x;
        if constexpr (GATHER) s_tok[threadIdx.x] = pair_tok[r];
        else                  s_tok[threadIdx.x] = r;
    }
    __syncthreads();

    const int wave = threadIdx.x >> 5;
    const int lane = threadIdx.x & 31;
    const int i0 = blockIdx.y * 256 + wave * 64;
    if (i0 >= ID) return;

    const int m = lane & 15;
    const int khalf = lane >> 4;           // lanes 0-15: K={0,1}, lanes 16-31: K={2,3}
    const int tok0 = s_tok[m];
    const int tok1 = s_tok[16 + m];
    // padded rows read a zeroed scratch row: branch-free inner loop
    const float* xrow0 = (tok0 >= 0) ? (X + (size_t)tok0 * HD) : zrow;
    const float* xrow1 = (tok1 >= 0) ? (X + (size_t)tok1 * HD) : zrow;
    const float* We = W + (size_t)e * ((size_t)HD * ID);
    const int col = i0 + m;

    v8f p0 = {}, p1 = {}, p2 = {}, p3 = {};   // rows 0..15
    v8f q0 = {}, q1 = {}, q2 = {}, q3 = {};   // rows 16..31
    for (int k0 = 0; k0 < HD; k0 += 4) {
        const int kk = k0 + khalf * 2;
        v2f a0, a1;
        a0.x = xrow0[kk]; a0.y = xrow0[kk + 1];
        a1.x = xrow1[kk]; a1.y = xrow1[kk + 1];
        const float* w0 = We + (size_t)kk * ID + col;
        const float* w1 = w0 + ID;
        v2f b;
        b.x = w0[0];  b.y = w1[0];
        p0 = wmma4(a0, b, p0); q0 = wmma4(a1, b, q0);
        b.x = w0[16]; b.y = w1[16];
        p1 = wmma4(a0, b, p1); q1 = wmma4(a1, b, q1);
        b.x = w0[32]; b.y = w1[32];
        p2 = wmma4(a0, b, p2); q2 = wmma4(a1, b, q2);
        b.x = w0[48]; b.y = w1[48];
        p3 = wmma4(a0, b, p3); q3 = wmma4(a1, b, q3);
        __builtin_prefetch(w0 + 4 * (size_t)ID, 0, 1);
    }

    const int mbase = khalf * 8;           // C rows: lanes 0-15 -> M=r, lanes 16-31 -> M=8+r
#pragma unroll
    for (int r = 0; r < 8; ++r) {
        const size_t row0 = (size_t)tile * ROWT + mbase + r;
        float* o0 = OUT + row0 * ID + col;
        o0[0]  = silu(p0[r]);
        o0[16] = silu(p1[r]);
        o0[32] = silu(p2[r]);
        o0[48] = silu(p3[r]);
        float* o1 = o0 + (size_t)16 * ID;  // rows 16..31 of the tile
        o1[0]  = silu(q0[r]);
        o1[16] = silu(q1[r]);
        o1[32] = silu(q2[r]);
        o1[48] = silu(q3[r]);
    }
}

// ---------------- fc2: Y[tok, h] (+)= w * ( H1[row,:] @ W[e][:, h] ) ----------------
template <bool SCATTER>
__global__ void fc2_kernel(const float* __restrict__ Hbuf, const float* __restrict__ W,
                           float* __restrict__ Y,
                           const int* __restrict__ tile_expert,
                           const int* __restrict__ pair_tok,
                           const float* __restrict__ pair_w) {
    __shared__ int   s_tok[ROWT];
    __shared__ float s_w[ROWT];
    const int tile = blockIdx.x;
    int e = 0;
    if constexpr (SCATTER) {
        e = tile_expert[tile];
        if (e < 0) return;
    }
    if (threadIdx.x < ROWT) {
        int r = tile * ROWT + threadIdx.x;
        if constexpr (SCATTER) { s_tok[threadIdx.x] = pair_tok[r]; s_w[threadIdx.x] = pair_w[r]; }
        else                   { s_tok[threadIdx.x] = r;           s_w[threadIdx.x] = 1.f; }
    }
    __syncthreads();

    const int wave = threadIdx.x >> 5;
    const int lane = threadIdx.x & 31;
    const int h0 = blockIdx.y * 256 + wave * 64;
    if (h0 >= HD) return;

    const int m = lane & 15;
    const int khalf = lane >> 4;
    const float* arow0 = Hbuf + ((size_t)tile * ROWT + m) * ID;
    const float* arow1 = arow0 + (size_t)16 * ID;
    const float* We = W + (size_t)e * ((size_t)ID * HD);
    const int col = h0 + m;

    v8f p0 = {}, p1 = {}, p2 = {}, p3 = {};
    v8f q0 = {}, q1 = {}, q2 = {}, q3 = {};
    for (int k0 = 0; k0 < ID; k0 += 4) {
        const int kk = k0 + khalf * 2;
        v2f a0, a1;
        a0.x = arow0[kk]; a0.y = arow0[kk + 1];
        a1.x = arow1[kk]; a1.y = arow1[kk + 1];
        const float* w0 = We + (size_t)kk * HD + col;
        const float* w1 = w0 + HD;
        v2f b;
        b.x = w0[0];  b.y = w1[0];
        p0 = wmma4(a0, b, p0); q0 = wmma4(a1, b, q0);
        b.x = w0[16]; b.y = w1[16];
        p1 = wmma4(a0, b, p1); q1 = wmma4(a1, b, q1);
        b.x = w0[32]; b.y = w1[32];
        p2 = wmma4(a0, b, p2); q2 = wmma4(a1, b, q2);
        b.x = w0[48]; b.y = w1[48];
        p3 = wmma4(a0, b, p3); q3 = wmma4(a1, b, q3);
        __builtin_prefetch(w0 + 4 * (size_t)HD, 0, 1);
    }

    const int mbase = khalf * 8;
#pragma unroll
    for (int r = 0; r < 8; ++r) {
        const int rowm = mbase + r;
        if constexpr (SCATTER) {
            int tok0 = s_tok[rowm];
            if (tok0 >= 0) {
                float w = s_w[rowm];
                float* yr = Y + (size_t)tok0 * HD + col;
                atomicAdd(yr + 0,  p0[r] * w);
                atomicAdd(yr + 16, p1[r] * w);
                atomicAdd(yr + 32, p2[r] * w);
                atomicAdd(yr + 48, p3[r] * w);
            }
            int tok1 = s_tok[16 + rowm];
            if (tok1 >= 0) {
                float w = s_w[16 + rowm];
                float* yr = Y + (size_t)tok1 * HD + col;
                atomicAdd(yr + 0,  q0[r] * w);
                atomicAdd(yr + 16, q1[r] * w);
                atomicAdd(yr + 32, q2[r] * w);
                atomicAdd(yr + 48, q3[r] * w);
            }
        } else {
            float* yr0 = Y + ((size_t)tile * ROWT + rowm) * HD + col;
            yr0[0]  = p0[r];
            yr0[16] = p1[r];
            yr0[32] = p2[r];
            yr0[48] = p3[r];
            float* yr1 = yr0 + (size_t)16 * HD;
            yr1[0]  = q0[r];
            yr1[16] = q1[r];
            yr1[32] = q2[r];
            yr1[48] = q3[r];
        }
    }
}

// ---------------- host launch ----------------
extern "C" void kernel_launch(void* const* d_in, const int* in_sizes, int n_in,
                              void* d_out, int out_size, void* d_ws, size_t ws_size,
                              hipStream_t stream) {
    const float* x   = (const float*)d_in[0];   // [NTOK, HD]
    const float* gw  = (const float*)d_in[1];   // [NE, HD]
    const float* W1  = (const float*)d_in[2];   // [NE, HD, ID]
    const float* W2  = (const float*)d_in[3];   // [NE, ID, HD]
    const float* Ws1 = (const float*)d_in[4];   // [HD, ID]
    const float* Ws2 = (const float*)d_in[5];   // [ID, HD]
    float* y = (float*)d_out;                   // [NTOK*HD] then aux at [NTOK*HD]

    char* ws = (char*)d_ws;
    int*   count    = (int*)  (ws + OFF_COUNT);
    int*   cursor   = (int*)  (ws + OFF_CURSOR);
    float* prob_sum = (float*)(ws + OFF_PROBSUM);
    int*   tile_exp = (int*)  (ws + OFF_TILEEXP);
    int*   top_idx  = (int*)  (ws + OFF_TOPIDX);
    float* top_val  = (float*)(ws + OFF_TOPVAL);
    int*   pair_tok = (int*)  (ws + OFF_PAIRTOK);
    float* pair_w   = (float*)(ws + OFF_PAIRW);
    float* zrow     = (float*)(ws + OFF_ZERO);
    float* h1p      = (float*)(ws + OFF_H1);    // [PAIR_CAP, ID]
    float* hs       = (float*)(ws + OFF_HS);    // [NTOK, ID]

    init_kernel<<<(PAIR_CAP + 255) / 256, 256, 0, stream>>>(count, cursor, prob_sum, pair_tok, zrow);
    router_kernel<<<NTOK / 4, 128, 0, stream>>>(x, gw, count, prob_sum, top_idx, top_val);
    build_offsets<<<1, 1, 0, stream>>>(count, cursor, tile_exp);
    scatter_kernel<<<NTOK / 256, 256, 0, stream>>>(top_idx, top_val, cursor, pair_tok, pair_w);
    aux_kernel<<<1, 1, 0, stream>>>(count, prob_sum, y + (size_t)NTOK * HD);

    // shared expert first (overwrites y), then routed experts accumulate
    fc1_kernel<false><<<dim3(NTOK / ROWT, (ID + 255) / 256), 128, 0, stream>>>(
        x, Ws1, hs, nullptr, nullptr, zrow);
    fc2_kernel<false><<<dim3(NTOK / ROWT, HD / 256), 128, 0, stream>>>(
        hs, Ws2, y, nullptr, nullptr, nullptr);

    fc1_kernel<true><<<dim3(MAX_TILES, (ID + 255) / 256), 128, 0, stream>>>(
        x, W1, h1p, tile_exp, pair_tok, zrow);
    fc2_kernel<true><<<dim3(MAX_TILES, HD / 256), 128, 0, stream>>>(
        h1p, W2, y, tile_exp, pair_tok, pair_w);
}